// AdvancedPartialAttentionMasking_60292750901348
// MI455X (gfx1250) — compile-verified
//
#include <hip/hip_runtime.h>

// Problem constants (from reference): x is [64, 256, 56, 56] f32
#define B_      64
#define C_      256
#define HW_     3136          // 56*56
#define HW4_    (HW_ / 4)     // 784 float4s per row
#define K_KEEP  128           // C * (1 - 0.5)
#define EPS_    1e-6f
#define NTHREADS 256
#define NWAVES  (NTHREADS / 32)

typedef float v4f __attribute__((ext_vector_type(4)));

// ---------------------------------------------------------------------------
// Kernel 1: per-(b,c) row entropy of (softmax + eps).
// One block per row. Row (12544 B) staged to LDS via CDNA5 async tensor-path
// (global_load_async_to_lds_b128, ASYNCcnt), then 3 LDS passes:
//   A: m = max(x)            B: S = sum exp(x-m), cache e in LDS
//   C: acc = sum q*log q, q = e/S + eps     -> entropy = -acc
// ---------------------------------------------------------------------------
__global__ __launch_bounds__(NTHREADS)
void entropy_kernel(const float* __restrict__ x, float* __restrict__ ent) {
    __shared__ float tile[HW_];       // 12544 bytes
    __shared__ float wred[NWAVES];    // cross-wave reduction scratch

    const int row  = blockIdx.x;      // b*C + c
    const int t    = threadIdx.x;
    const int lane = t & 31;
    const int wave = t >> 5;
    const float* base = x + (size_t)row * HW_;

    // ---- async stage global row -> LDS (16 bytes per lane per issue) ----
    for (int i4 = t; i4 < HW4_; i4 += NTHREADS) {
        unsigned int       loff = (unsigned int)(size_t)(&tile[i4 * 4]); // LDS byte offset (low 32 bits of flat addr)
        unsigned long long ga   = (unsigned long long)(size_t)(base + i4 * 4);
        asm volatile("global_load_async_to_lds_b128 %0, %1, off"
                     :: "v"(loff), "v"(ga) : "memory");
    }
    asm volatile("s_wait_asynccnt 0" ::: "memory");  // this wave's async writes to LDS done
    __syncthreads();                                 // make all waves' data visible

    // ---- pass A: row max ----
    float m = -3.402823466e38f;
    for (int i = t; i < HW_; i += NTHREADS) m = fmaxf(m, tile[i]);
    #pragma unroll
    for (int o = 16; o > 0; o >>= 1) m = fmaxf(m, __shfl_xor(m, o, 32));
    if (lane == 0) wred[wave] = m;
    __syncthreads();
    float bm = wred[0];
    #pragma unroll
    for (int w = 1; w < NWAVES; ++w) bm = fmaxf(bm, wred[w]);
    __syncthreads();   // everyone has bm; wred free for reuse

    // ---- pass B: S = sum exp(x - m); cache e back into tile ----
    float s = 0.0f;
    for (int i = t; i < HW_; i += NTHREADS) {
        float e = __expf(tile[i] - bm);
        tile[i] = e;                 // same indices this thread read: no cross-thread hazard
        s += e;
    }
    #pragma unroll
    for (int o = 16; o > 0; o >>= 1) s += __shfl_xor(s, o, 32);
    if (lane == 0) wred[wave] = s;
    __syncthreads();
    float S = 0.0f;
    #pragma unroll
    for (int w = 0; w < NWAVES; ++w) S += wred[w];
    const float invS = 1.0f / S;
    __syncthreads();   // wred free for reuse

    // ---- pass C: acc = sum q * log q, q = p + eps ----
    float acc = 0.0f;
    for (int i = t; i < HW_; i += NTHREADS) {
        float q = tile[i] * invS + EPS_;
        acc += q * __logf(q);
    }
    #pragma unroll
    for (int o = 16; o > 0; o >>= 1) acc += __shfl_xor(acc, o, 32);
    if (lane == 0) wred[wave] = acc;
    __syncthreads();
    if (t == 0) {
        float a = 0.0f;
        #pragma unroll
        for (int w = 0; w < NWAVES; ++w) a += wred[w];
        ent[row] = -a;               // entropy
    }
}

// ---------------------------------------------------------------------------
// Kernel 2: per-batch top-k mask from entropies.
// importance = -entropy; jax top_k tie-break = lower index wins.
// rank(c) = #{ j : ent[j] < ent[c]  ||  (ent[j] == ent[c] && j < c) }
// ---------------------------------------------------------------------------
__global__ __launch_bounds__(C_)
void mask_kernel(const float* __restrict__ ent, float* __restrict__ mask) {
    __shared__ float se[C_];
    const int b = blockIdx.x;
    const int c = threadIdx.x;
    const float e = ent[b * C_ + c];
    se[c] = e;
    __syncthreads();
    int rank = 0;
    #pragma unroll 8
    for (int j = 0; j < C_; ++j) {
        float ej = se[j];
        rank += (ej < e) || (ej == e && j < c);
    }
    mask[b * C_ + c] = (rank < K_KEEP) ? 1.0f : 0.0f;
}

// ---------------------------------------------------------------------------
// Kernel 3: out = x * mask[channel], float4 streaming with NT hints.
// 784 float4s per channel row -> channel id = i4 / 784.
// ---------------------------------------------------------------------------
__global__ __launch_bounds__(NTHREADS)
void apply_kernel(const float* __restrict__ x, const float* __restrict__ mask,
                  float* __restrict__ out, int n4) {
    const int stride = gridDim.x * blockDim.x;
    for (int i = blockIdx.x * blockDim.x + threadIdx.x; i < n4; i += stride) {
        unsigned int rowc = (unsigned int)i / (unsigned int)HW4_;  // b*C + c
        float mval = mask[rowc];
        v4f v = __builtin_nontemporal_load((const v4f*)x + i);
        v4f r = v * mval;
        __builtin_nontemporal_store(r, (v4f*)out + i);
    }
}

// ---------------------------------------------------------------------------
extern "C" void kernel_launch(void* const* d_in, const int* in_sizes, int n_in,
                              void* d_out, int out_size, void* d_ws, size_t ws_size,
                              hipStream_t stream) {
    const float* x = (const float*)d_in[0];
    float* out = (float*)d_out;

    // workspace: [0, 64KB) entropy, [64KB, 128KB) mask
    float* ent  = (float*)d_ws;
    float* mask = ent + (size_t)B_ * C_;

    entropy_kernel<<<B_ * C_, NTHREADS, 0, stream>>>(x, ent);
    mask_kernel<<<B_, C_, 0, stream>>>(ent, mask);

    const int n4 = B_ * C_ * HW4_;                 // 12,845,056 float4s
    int blocks = (n4 + NTHREADS - 1) / NTHREADS;   // one float4 per thread
    apply_kernel<<<blocks, NTHREADS, 0, stream>>>(x, mask, out, n4);
}